// SelfAttentionLayer_55379308314657
// MI455X (gfx1250) — compile-verified
//
#include <hip/hip_runtime.h>
#include <hip/hip_bf16.h>
#include <math.h>

// Self-attention layer for MI455X (gfx1250), fp32 end-to-end via
// V_WMMA_F32_16X16X4_F32. Flash-style online softmax (no 256MB score buffer).
// Register-blocked GEMMs: one wave owns a 64x64 (or 32x64) macro-tile so each
// A/B fragment load feeds 4 WMMAs.

typedef __attribute__((ext_vector_type(2))) float v2f;
typedef __attribute__((ext_vector_type(8))) float v8f;

#define S_LEN   2048
#define DMODEL  1024
#define DK      64
#define NH      16

static __device__ __forceinline__ v8f wmma_f32(v2f a, v2f b, v8f c) {
  // 8 args: (neg_a, A, neg_b, B, c_mod, C, reuse_a, reuse_b)
  return __builtin_amdgcn_wmma_f32_16x16x4_f32(false, a, false, b, (short)0, c,
                                               false, false);
}

// ---------------------------------------------------------------------------
// Kernel A: Q/K/V projections.  q[h,s,k] = sum_d x[s,d] * w[h,d,k]
// One wave computes a 64x64 macro-tile (4x4 WMMA tiles): per k-step of 4,
// 4 b64 + 8 b32 loads feed 16 WMMAs.
// Tiles: 3 matrices * 16 heads * 32 m-blocks = 1536 waves.
// ---------------------------------------------------------------------------
__global__ void __launch_bounds__(128)
qkv_proj_kernel(const float* __restrict__ x,  const float* __restrict__ wq,
                const float* __restrict__ wk, const float* __restrict__ wv,
                float* __restrict__ qb, float* __restrict__ kb,
                float* __restrict__ vb)
{
  const int wid  = threadIdx.x >> 5;
  const int lane = threadIdx.x & 31;
  int t = blockIdx.x * 4 + wid;
  const int mt = t & 31; t >>= 5;
  const int which = t % 3;
  const int h = t / 3;

  const float* W = (which == 0 ? wq : which == 1 ? wk : wv)
                   + (size_t)h * DMODEL * DK;
  float* O = (which == 0 ? qb : which == 1 ? kb : vb)
             + (size_t)h * S_LEN * DK;

  const int col  = lane & 15;       // A-frag row / B-frag col
  const int half = lane >> 4;

  const float* Abase = x + (size_t)(mt * 64 + col) * DMODEL + half * 2;
  const float* Bbase = W + col;

  v8f acc[4][4];
#pragma unroll
  for (int mi = 0; mi < 4; ++mi)
#pragma unroll
    for (int ni = 0; ni < 4; ++ni) acc[mi][ni] = (v8f){};

#pragma unroll 2
  for (int k0 = 0; k0 < DMODEL; k0 += 4) {
    const int kbi = k0 + half * 2;
    v2f a[4];
#pragma unroll
    for (int mi = 0; mi < 4; ++mi)
      a[mi] = *(const v2f*)(Abase + (size_t)mi * 16 * DMODEL + k0);
    v2f b[4];
#pragma unroll
    for (int ni = 0; ni < 4; ++ni) {
      b[ni].x = Bbase[(size_t)kbi * DK + ni * 16];
      b[ni].y = Bbase[(size_t)(kbi + 1) * DK + ni * 16];
    }
#pragma unroll
    for (int mi = 0; mi < 4; ++mi)
#pragma unroll
      for (int ni = 0; ni < 4; ++ni)
        acc[mi][ni] = wmma_f32(a[mi], b[ni], acc[mi][ni]);
  }

  const int rb = half * 8;
#pragma unroll
  for (int mi = 0; mi < 4; ++mi)
#pragma unroll
    for (int ni = 0; ni < 4; ++ni)
#pragma unroll
      for (int r = 0; r < 8; ++r)
        O[(size_t)(mt * 64 + mi * 16 + rb + r) * DK + ni * 16 + col] =
            acc[mi][ni][r];
}

// ---------------------------------------------------------------------------
// Kernel B: flash attention per head.  One wave owns 16 query rows x d_v=64.
// Streams 128 key tiles of 16; online softmax; P bounced through LDS to
// convert C/D layout -> A-frag layout for the P x V WMMAs.
// ---------------------------------------------------------------------------
__global__ void __launch_bounds__(128)
attn_kernel(const float* __restrict__ qbuf, const float* __restrict__ kbuf,
            const float* __restrict__ vbuf, float* __restrict__ zb)
{
  __shared__ float ps[4][16 * 16];               // 1KB per wave P-staging

  const int wid  = threadIdx.x >> 5;
  const int lane = threadIdx.x & 31;
  const int t  = blockIdx.x * 4 + wid;
  const int mt = t & 127;
  const int h  = t >> 7;

  const int col  = lane & 15;
  const int half = lane >> 4;
  const int rb   = half * 8;

  // Pin the Q tile (16 rows x 64) in registers as 16 A-fragments.
  const float* Qrow = qbuf + ((size_t)h * S_LEN + mt * 16 + col) * DK + half * 2;
  v2f qa[16];
#pragma unroll
  for (int ks = 0; ks < 16; ++ks)
    qa[ks] = *(const v2f*)(Qrow + ks * 4);

  v8f z0 = {}, z1 = {}, z2 = {}, z3 = {};
  float m[8], l[8];
#pragma unroll
  for (int r = 0; r < 8; ++r) { m[r] = -INFINITY; l[r] = 0.0f; }

  float* psw = ps[wid];
  const float* Khead = kbuf + ((size_t)h * S_LEN + col) * DK + half * 2;
  const float* Vhead = vbuf + (size_t)h * S_LEN * DK;

  for (int kt = 0; kt < S_LEN / 16; ++kt) {
    // Prefetch next key/value tiles (global_prefetch_b8); stays in workspace.
    __builtin_prefetch(Khead + (size_t)(kt + 1) * 16 * DK, 0, 3);
    __builtin_prefetch(Vhead + (size_t)((kt + 1) * 16 + col) * DK, 0, 3);

    // ---- S = Q K^T  (16x16, K-dim = 64 -> 16 WMMA steps) ----
    v8f s = {};
    const float* Kc = Khead + (size_t)kt * 16 * DK;
#pragma unroll
    for (int ks = 0; ks < 16; ++ks) {
      v2f b = *(const v2f*)(Kc + ks * 4);        // K[key=col][kb..kb+1]
      s = wmma_f32(qa[ks], b, s);
    }

    // ---- online softmax over the 16 new keys ----
    float corr[8];
#pragma unroll
    for (int r = 0; r < 8; ++r) {
      float sv = s[r] * 0.125f;                  // 1/sqrt(64)
      float mx = sv;
      mx = fmaxf(mx, __shfl_xor(mx, 1));
      mx = fmaxf(mx, __shfl_xor(mx, 2));
      mx = fmaxf(mx, __shfl_xor(mx, 4));
      mx = fmaxf(mx, __shfl_xor(mx, 8));         // row max (16-lane group)
      float mn = fmaxf(m[r], mx);
      float cr = __expf(m[r] - mn);
      float p  = __expf(sv - mn);
      float rs = p;
      rs += __shfl_xor(rs, 1);
      rs += __shfl_xor(rs, 2);
      rs += __shfl_xor(rs, 4);
      rs += __shfl_xor(rs, 8);                   // row sum
      l[r] = l[r] * cr + rs;
      m[r] = mn;
      corr[r] = cr;
      psw[(rb + r) * 16 + col] = p;              // P tile -> LDS (C layout)
    }
#pragma unroll
    for (int r = 0; r < 8; ++r) {
      z0[r] *= corr[r]; z1[r] *= corr[r];
      z2[r] *= corr[r]; z3[r] *= corr[r];
    }

    // ---- Z += P x V  (K-dim = 16 -> 4 WMMA steps x 4 n-tiles) ----
    const float* Vb = Vhead + (size_t)kt * 16 * DK;
#pragma unroll
    for (int ks = 0; ks < 4; ++ks) {
      const int kbi = ks * 4 + half * 2;
      v2f pa = *(const v2f*)(psw + col * 16 + kbi);  // A-frag of P from LDS
      const float* vp = Vb + (size_t)kbi * DK;
      v2f b0, b1, b2, b3;
      b0.x = vp[col];          b0.y = vp[DK + col];
      b1.x = vp[16 + col];     b1.y = vp[DK + 16 + col];
      b2.x = vp[32 + col];     b2.y = vp[DK + 32 + col];
      b3.x = vp[48 + col];     b3.y = vp[DK + 48 + col];
      z0 = wmma_f32(pa, b0, z0);
      z1 = wmma_f32(pa, b1, z1);
      z2 = wmma_f32(pa, b2, z2);
      z3 = wmma_f32(pa, b3, z3);
    }
  }

  // ---- finalize: divide by row sums, store in [s, (h v)] layout ----
#pragma unroll
  for (int r = 0; r < 8; ++r) {
    float inv = 1.0f / l[r];
    z0[r] *= inv; z1[r] *= inv; z2[r] *= inv; z3[r] *= inv;
  }
  float* Z = zb + (size_t)(mt * 16 + rb) * (NH * DK) + h * DK;
#pragma unroll
  for (int r = 0; r < 8; ++r) {
    Z[(size_t)r * NH * DK + col]      = z0[r];
    Z[(size_t)r * NH * DK + 16 + col] = z1[r];
    Z[(size_t)r * NH * DK + 32 + col] = z2[r];
    Z[(size_t)r * NH * DK + 48 + col] = z3[r];
  }
}

// ---------------------------------------------------------------------------
// Kernel C: output projection  out[2048,1024] = Z[2048,1024] @ wo[1024,1024]
// One wave computes a 32x64 macro-tile (2x4 WMMA tiles): per k-step of 4,
// 2 b64 + 8 b32 loads feed 8 WMMAs.  64 m-blocks * 16 n-blocks = 1024 waves.
// ---------------------------------------------------------------------------
__global__ void __launch_bounds__(128)
out_proj_kernel(const float* __restrict__ zb, const float* __restrict__ wo,
                float* __restrict__ out)
{
  const int wid  = threadIdx.x >> 5;
  const int lane = threadIdx.x & 31;
  const int t  = blockIdx.x * 4 + wid;
  const int nt = t & 15;
  const int mt = t >> 4;

  const int col  = lane & 15;
  const int half = lane >> 4;

  const float* Abase = zb + (size_t)(mt * 32 + col) * DMODEL + half * 2;
  const float* Bbase = wo + nt * 64 + col;

  v8f acc[2][4];
#pragma unroll
  for (int mi = 0; mi < 2; ++mi)
#pragma unroll
    for (int ni = 0; ni < 4; ++ni) acc[mi][ni] = (v8f){};

#pragma unroll 2
  for (int k0 = 0; k0 < DMODEL; k0 += 4) {
    const int kbi = k0 + half * 2;
    v2f a[2];
#pragma unroll
    for (int mi = 0; mi < 2; ++mi)
      a[mi] = *(const v2f*)(Abase + (size_t)mi * 16 * DMODEL + k0);
    v2f b[4];
#pragma unroll
    for (int ni = 0; ni < 4; ++ni) {
      b[ni].x = Bbase[(size_t)kbi * DMODEL + ni * 16];
      b[ni].y = Bbase[(size_t)(kbi + 1) * DMODEL + ni * 16];
    }
#pragma unroll
    for (int mi = 0; mi < 2; ++mi)
#pragma unroll
      for (int ni = 0; ni < 4; ++ni)
        acc[mi][ni] = wmma_f32(a[mi], b[ni], acc[mi][ni]);
  }

  const int rb = half * 8;
#pragma unroll
  for (int mi = 0; mi < 2; ++mi)
#pragma unroll
    for (int ni = 0; ni < 4; ++ni)
#pragma unroll
      for (int r = 0; r < 8; ++r)
        out[(size_t)(mt * 32 + mi * 16 + rb + r) * DMODEL
            + nt * 64 + ni * 16 + col] = acc[mi][ni][r];
}

// ---------------------------------------------------------------------------
extern "C" void kernel_launch(void* const* d_in, const int* in_sizes, int n_in,
                              void* d_out, int out_size, void* d_ws,
                              size_t ws_size, hipStream_t stream) {
  const float* x  = (const float*)d_in[0];
  const float* wq = (const float*)d_in[1];
  const float* wk = (const float*)d_in[2];
  const float* wv = (const float*)d_in[3];
  const float* wo = (const float*)d_in[4];
  float* out = (float*)d_out;

  // Workspace: Q, K, V (8MB each) + Z (8MB) = 32MB, fully L2-resident.
  float* qb = (float*)d_ws;
  float* kb = qb + (size_t)NH * S_LEN * DK;
  float* vb = kb + (size_t)NH * S_LEN * DK;
  float* zb = vb + (size_t)NH * S_LEN * DK;

  // 3 matrices * 16 heads * 32 m-blocks = 1536 waves / 4 per block
  qkv_proj_kernel<<<384, 128, 0, stream>>>(x, wq, wk, wv, qb, kb, vb);
  // 16 heads * 128 m-tiles = 2048 waves
  attn_kernel<<<512, 128, 0, stream>>>(qb, kb, vb, zb);
  // 64 m-blocks * 16 n-blocks = 1024 waves
  out_proj_kernel<<<256, 128, 0, stream>>>(zb, wo, out);
}